// GaussianImage_89739046683037
// MI455X (gfx1250) — compile-verified
//
#include <hip/hip_runtime.h>
#include <math.h>

typedef __attribute__((ext_vector_type(2))) float v2f;
typedef __attribute__((ext_vector_type(8))) float v8f;

#define N_GAUSS 128
#define PLANE   512
#define PI_F    3.14159265358979323846f

// ---------------------------------------------------------------------------
// Kernel 1: per-gaussian quadratic coefficients (8 floats each, padded K=8).
//   expo_n(x,y) = c0*x^2 + c1*x*y + c2*y^2 + c3*x + c4*y + c5  (c6=c7=0)
// with -0.5 * diff^T inv(cov) diff folded in analytically:
//   inv(cov) = R diag(1/s0^2, 1/s1^2) R^T,  R = [[c,-s],[s,c]]
// ---------------------------------------------------------------------------
__global__ void ga_coeff_kernel(const float* __restrict__ mean,
                                const float* __restrict__ scale,
                                const float* __restrict__ theta,
                                float* __restrict__ coef) {
  int n = threadIdx.x;
  if (n >= N_GAUSS) return;
  float mx = mean[2 * n + 0], my = mean[2 * n + 1];
  float s0 = scale[2 * n + 0], s1 = scale[2 * n + 1];
  float th = (1.0f + sinf(theta[n])) * PI_F;
  float ct = cosf(th), st = sinf(th);
  float i0 = 1.0f / (s0 * s0);
  float i1 = 1.0f / (s1 * s1);
  // inv(cov) = [[A,B],[B,C]]
  float A = ct * ct * i0 + st * st * i1;
  float B = ct * st * (i0 - i1);
  float C = st * st * i0 + ct * ct * i1;
  // q(x,y) = A x^2 + 2B xy + C y^2 + Dx x + Ey y + F   (diff = p - mean)
  float Dx = -2.0f * (A * mx + B * my);
  float Ey = -2.0f * (B * mx + C * my);
  float F  = A * mx * mx + 2.0f * B * mx * my + C * my * my;
  float* c8 = coef + n * 8;
  c8[0] = -0.5f * A;        // * x^2
  c8[1] = -1.0f * B;        // * x*y   (-0.5 * 2B)
  c8[2] = -0.5f * C;        // * y^2
  c8[3] = -0.5f * Dx;       // * x
  c8[4] = -0.5f * Ey;       // * y
  c8[5] = -0.5f * F;        // * 1
  c8[6] = 0.0f;
  c8[7] = 0.0f;
}

// ---------------------------------------------------------------------------
// Kernel 2: one wave32 per 16-pixel tile. Exponents via
// V_WMMA_F32_16X16X4_F32 (M=16 pixels, N=16 gaussians, 2 K-steps of 4),
// then v_exp_f32 + per-row shfl_xor reduction over N (the 16-lane halves).
// ---------------------------------------------------------------------------
__global__ void __launch_bounds__(256) ga_splat_kernel(
    const float* __restrict__ pix,    // (512,512,2)
    const float* __restrict__ coef,   // (128,8)
    const float* __restrict__ rgb,    // (128,3)
    const float* __restrict__ alpha,  // (128,1)
    float* __restrict__ out) {        // (512,512,3)
  const int lane = threadIdx.x & 31;
  const int wid  = (int)((blockIdx.x * blockDim.x + threadIdx.x) >> 5);
  const int m    = lane & 15;   // pixel row of A-matrix / gaussian col of B
  const int half = lane >> 4;   // selects K=0,1 vs K=2,3 slots per ISA layout
  const int pixBase = wid * 16; // 16 consecutive pixels, same image row

  const int p = pixBase + m;
  const float x = pix[2 * p + 0];
  const float y = pix[2 * p + 1];

  const float aN = alpha[N_GAUSS - 1];
  const float K0 = rgb[(N_GAUSS - 1) * 3 + 0] * aN;
  const float K1 = rgb[(N_GAUSS - 1) * 3 + 1] * aN;
  const float K2 = rgb[(N_GAUSS - 1) * 3 + 2] * aN;

  // A matrix (16x4 f32): VGPR0 = K0/K2, VGPR1 = K1/K3 split by lane half.
  // Features: [x^2, xy, y^2, x, y, 1, 0, 0]
  v2f a0, a1;
  a0[0] = half ? y * y : x * x;   // k-step0: K=0 (x^2) | K=2 (y^2)
  a0[1] = half ? x     : x * y;   //          K=1 (xy)  | K=3 (x)
  a1[0] = half ? 0.0f  : y;       // k-step1: K=4 (y)   | K=6 (0)
  a1[1] = half ? 0.0f  : 1.0f;    //          K=5 (1)   | K=7 (0)

  // Preload B operands for all 8 gaussian tiles (lane owns gaussian t*16+m).
  float bt[8][4];
#pragma unroll
  for (int t = 0; t < 8; ++t) {
    const float* c8 = coef + (t * 16 + m) * 8;
    bt[t][0] = c8[half ? 2 : 0];
    bt[t][1] = c8[half ? 3 : 1];
    bt[t][2] = half ? 0.0f : c8[4];
    bt[t][3] = half ? 0.0f : c8[5];
  }

  float sums[8] = {0.f, 0.f, 0.f, 0.f, 0.f, 0.f, 0.f, 0.f};

#pragma unroll
  for (int t = 0; t < 8; ++t) {
    v2f b0, b1;
    b0[0] = bt[t][0]; b0[1] = bt[t][1];
    b1[0] = bt[t][2]; b1[1] = bt[t][3];
    v8f acc = {0.f, 0.f, 0.f, 0.f, 0.f, 0.f, 0.f, 0.f};
    acc = __builtin_amdgcn_wmma_f32_16x16x4_f32(
        false, a0, false, b0, (short)0, acc, false, false);
    acc = __builtin_amdgcn_wmma_f32_16x16x4_f32(
        false, a1, false, b1, (short)0, acc, false, false);
#pragma unroll
    for (int r = 0; r < 8; ++r) sums[r] += __expf(acc[r]);
  }

  // Reduce over N (16 gaussians) within each 16-lane half, per C/D row r.
  // VGPR r holds pixel M=r (lanes 0-15) and M=r+8 (lanes 16-31).
#pragma unroll
  for (int r = 0; r < 8; ++r) {
    float v = sums[r];
    v += __shfl_xor(v, 8, 32);
    v += __shfl_xor(v, 4, 32);
    v += __shfl_xor(v, 2, 32);
    v += __shfl_xor(v, 1, 32);
    if (m == r) {
      int pp = pixBase + r + half * 8;
      out[3 * pp + 0] = K0 * v;
      out[3 * pp + 1] = K1 * v;
      out[3 * pp + 2] = K2 * v;
    }
  }
}

// ---------------------------------------------------------------------------
extern "C" void kernel_launch(void* const* d_in, const int* in_sizes, int n_in,
                              void* d_out, int out_size, void* d_ws, size_t ws_size,
                              hipStream_t stream) {
  const float* mean  = (const float*)d_in[0];
  const float* rgb   = (const float*)d_in[1];
  const float* alpha = (const float*)d_in[2];
  const float* scale = (const float*)d_in[3];
  const float* theta = (const float*)d_in[4];
  const float* pix   = (const float*)d_in[5];
  float* out  = (float*)d_out;
  float* coef = (float*)d_ws;  // 128*8 floats = 4 KB

  ga_coeff_kernel<<<1, 128, 0, stream>>>(mean, scale, theta, coef);

  const int tiles   = (PLANE * PLANE) / 16;      // 16384 waves, 1 tile each
  const int threads = 256;                       // 8 wave32 per block
  const int blocks  = tiles * 32 / threads;      // 2048
  ga_splat_kernel<<<blocks, threads, 0, stream>>>(pix, coef, rgb, alpha, out);
}